// TransformerEmbedding_661424963603
// MI455X (gfx1250) — compile-verified
//
#include <hip/hip_runtime.h>

// ---------------------------------------------------------------------------
// Problem constants (match reference)
// ---------------------------------------------------------------------------
#define N_KINGS 27
#define VOCAB   32000
#define EDIM    256
#define DDIM    512
#define BATCH   16
#define SEQ     512
#define EPS_F   1e-5f

// LDS row stride: 256 + 4 pad floats. With b64 LDS loads the 32 lanes of a
// wave touch 64 distinct banks (4*m + {0..3} mod 64) -> conflict-free.
#define LDS_STRIDE 260

typedef float v2f __attribute__((ext_vector_type(2)));
typedef float v8f __attribute__((ext_vector_type(8)));

// D = A(16xK) * B(Kx16) + C via V_WMMA_F32_16X16X4_F32, K-chained on acc.
__device__ __forceinline__ v8f wmma_f32_k4(v2f a, v2f b, v8f acc) {
  return __builtin_amdgcn_wmma_f32_16x16x4_f32(
      /*neg_a=*/false, a, /*neg_b=*/false, b,
      /*c_mod=*/(short)0, acc, /*reuse_a=*/false, /*reuse_b=*/false);
}

// ---------------------------------------------------------------------------
// Stage 1:  T[b][e][d] = sum_f W[d][e][f] * king_table[kid[b]][f]
//   A = kemb  (M = batch = 16, K = f)
//   B = 16 consecutive d-rows of W at fixed e (N = d, K = f)
//   -> W is streamed exactly once (M spans all 16 batches),
//      each lane reads its own contiguous W row, T stores are coalesced.
// W loads are NON-TEMPORAL: 134 MB one-shot stream must not evict T (8 MB,
// re-read ~32x by stage 2) from the 192 MB L2.
// 8192 tiles = 256 e-values * 32 d-tiles; 1 wave per tile, 8 waves/block.
// ---------------------------------------------------------------------------
__global__ __launch_bounds__(256) void king_w_gemm(
    const float* __restrict__ Wm,          // [D][E][E]
    const float* __restrict__ king_table,  // [N_KINGS][E]
    const int*   __restrict__ king_id,     // [B]
    float*       __restrict__ T)           // [B][E][D]  (workspace)
{
  __shared__ float kemb[BATCH * LDS_STRIDE];

  const int tid = threadIdx.x;
  // Cooperative load: kemb[b][f] = king_table[kid[b]][f]   (16 x 256 f32)
  for (int idx = tid; idx < BATCH * EDIM; idx += 256) {
    const int n = idx >> 8;            // batch
    const int f = idx & (EDIM - 1);
    const int kid = king_id[n];
    kemb[n * LDS_STRIDE + f] = king_table[kid * EDIM + f];
  }
  __syncthreads();

  const int wave = tid >> 5;
  const int lane = tid & 31;
  const int tile = blockIdx.x * 8 + wave;      // 1024 blocks * 8 = 8192 tiles
  const int e    = tile >> 5;                  // 0..255
  const int d0   = (tile & 31) << 4;           // 0,16,...,496
  const int ln   = lane & 15;                  // M (A) / N (B) index
  const int kOff = (lane >> 4) << 1;           // 0 or 2 (K sub-pair)

  // B operand: lane ln streams W row (d0+ln, e, :) ; 8B every 4 K-values.
  const float* wrow = Wm + ((size_t)(d0 + ln) * EDIM + e) * EDIM + kOff;
  // A operand: lane ln reads kemb row ln.
  const float* arow = kemb + ln * LDS_STRIDE + kOff;

  v8f acc = {};
#pragma unroll 8
  for (int k0 = 0; k0 < EDIM; k0 += 4) {
    v2f a = *(const v2f*)(arow + k0);                         // ds_load_b64
    v2f b = __builtin_nontemporal_load((const v2f*)(wrow + k0)); // NT stream
    acc = wmma_f32_k4(a, b, acc);
  }

  // D layout: lane<16: vgpr r = D[M=r, N=lane]; lane>=16: D[M=8+r, N=lane-16]
  const int mBase = (lane >> 4) << 3;
#pragma unroll
  for (int r = 0; r < 8; ++r) {
    const int bi = mBase + r;                  // batch (M)
    T[((size_t)bi * EDIM + e) * DDIM + d0 + ln] = acc[r];   // coalesced 64B
  }
}

// ---------------------------------------------------------------------------
// Stage 2:  out[b][s][d] = sum_e tables[kid[b]][seq[b][s]][e] * T[b][e][d] + bias[d]
// One block per (batch, 16-row s-tile): gather seq tile into LDS once.
// Each wave owns 4 consecutive d-tiles, processed TWO AT A TIME with
// independent accumulators: each A operand (LDS) feeds two WMMA chains,
// halving DS traffic and hiding the WMMA->WMMA accumulation latency.
// B operand loads from T are coalesced across lanes (consecutive d).
// ---------------------------------------------------------------------------
__global__ __launch_bounds__(256) void seq_t_gemm(
    const float* __restrict__ tables,     // [N_KINGS][VOCAB][E]
    const int*   __restrict__ sequence,   // [B][S]
    const int*   __restrict__ king_id,    // [B]
    const float* __restrict__ T,          // [B][E][D]
    const float* __restrict__ bias,       // [D]
    float*       __restrict__ out)        // [B][S][D]
{
  __shared__ float seqT[16 * LDS_STRIDE];

  const int tid = threadIdx.x;
  const int blk = blockIdx.x;               // 512 = B * S/16
  const int b   = blk >> 5;
  const int s0  = (blk & 31) << 4;
  const int kid = king_id[b];

  // Gather 16 embedding rows (float4 vectorized). Table row 0 is zero in the
  // input, matching the reference's tables.at[:,0,:].set(0).
  for (int idx = tid; idx < 16 * (EDIM / 4); idx += 256) {
    const int i  = idx >> 6;                // s-row 0..15
    const int f4 = idx & 63;                // float4 index within row
    const int tok = sequence[b * SEQ + s0 + i];
    const float4 v =
        ((const float4*)(tables + ((size_t)kid * VOCAB + tok) * EDIM))[f4];
    float* dst = seqT + i * LDS_STRIDE + f4 * 4;
    dst[0] = v.x; dst[1] = v.y; dst[2] = v.z; dst[3] = v.w;
  }
  __syncthreads();

  const int wave = tid >> 5;
  const int lane = tid & 31;
  const int ln   = lane & 15;
  const int kOff = (lane >> 4) << 1;
  const float* arow = seqT + ln * LDS_STRIDE + kOff;   // A: seq tile (M = s)
  const int   mBase = (lane >> 4) << 3;

#pragma unroll
  for (int pass = 0; pass < 2; ++pass) {               // uniform per wave
    const int d0 = ((wave << 2) + (pass << 1)) << 4;   // wave owns 4 d-tiles
    // B: T[b][e = k][d0 + ln]; per K-step two b32 loads per tile, 64B coalesced.
    const float* brow0 = T + ((size_t)b * EDIM + kOff) * DDIM + d0 + ln;
    const float* brow1 = brow0 + 16;

    v8f acc0 = {};
    v8f acc1 = {};
#pragma unroll 8
    for (int k0 = 0; k0 < EDIM; k0 += 4) {
      v2f a = *(const v2f*)(arow + k0);
      v2f b0, b1;
      b0.x = brow0[(size_t)k0 * DDIM];
      b0.y = brow0[(size_t)(k0 + 1) * DDIM];
      b1.x = brow1[(size_t)k0 * DDIM];
      b1.y = brow1[(size_t)(k0 + 1) * DDIM];
      acc0 = wmma_f32_k4(a, b0, acc0);   // two independent WMMA chains:
      acc1 = wmma_f32_k4(a, b1, acc1);   // hides D->C accumulation latency
    }

    const float bb0 = bias[d0 + ln];
    const float bb1 = bias[d0 + 16 + ln];
#pragma unroll
    for (int r = 0; r < 8; ++r) {
      const int s = s0 + mBase + r;
      float* orow = out + ((size_t)b * SEQ + s) * DDIM + d0 + ln;
      orow[0]  = acc0[r] + bb0;          // coalesced 64B per half-wave
      orow[16] = acc1[r] + bb1;
    }
  }
}

// ---------------------------------------------------------------------------
// Stage 3: in-place LayerNorm over D=512. One wave32 per row; 4xfloat4 per
// lane; wave reduction via shuffles. Deterministic: stage 2 rewrites every
// element of d_out each call before this runs.
// ---------------------------------------------------------------------------
__global__ __launch_bounds__(256) void layernorm_inplace(
    float* __restrict__ out,
    const float* __restrict__ gamma,
    const float* __restrict__ beta)
{
  const int wave = threadIdx.x >> 5;
  const int lane = threadIdx.x & 31;
  const size_t row = (size_t)blockIdx.x * 8 + wave;    // 8192 rows
  float* p = out + row * DDIM;

  float4 v[4];
  float sum = 0.f, sq = 0.f;
#pragma unroll
  for (int i = 0; i < 4; ++i) {
    v[i] = ((const float4*)p)[lane + i * 32];
    sum += v[i].x + v[i].y + v[i].z + v[i].w;
    sq  += v[i].x * v[i].x + v[i].y * v[i].y + v[i].z * v[i].z + v[i].w * v[i].w;
  }
#pragma unroll
  for (int m = 16; m >= 1; m >>= 1) {
    sum += __shfl_xor(sum, m, 32);
    sq  += __shfl_xor(sq,  m, 32);
  }
  const float mean = sum * (1.0f / DDIM);
  const float var  = sq  * (1.0f / DDIM) - mean * mean;
  const float inv  = rsqrtf(var + EPS_F);

#pragma unroll
  for (int i = 0; i < 4; ++i) {
    const int q = lane + i * 32;
    const float4 g  = ((const float4*)gamma)[q];
    const float4 bt = ((const float4*)beta)[q];
    float4 r;
    r.x = (v[i].x - mean) * inv * g.x + bt.x;
    r.y = (v[i].y - mean) * inv * g.y + bt.y;
    r.z = (v[i].z - mean) * inv * g.z + bt.z;
    r.w = (v[i].w - mean) * inv * g.w + bt.w;
    ((float4*)p)[q] = r;
  }
}

// ---------------------------------------------------------------------------
extern "C" void kernel_launch(void* const* d_in, const int* in_sizes, int n_in,
                              void* d_out, int out_size, void* d_ws, size_t ws_size,
                              hipStream_t stream) {
  (void)in_sizes; (void)n_in; (void)out_size; (void)ws_size;

  const float* tables     = (const float*)d_in[0];  // [27][32000][256]
  const float* king_table = (const float*)d_in[1];  // [27][256]
  const float* Wm         = (const float*)d_in[2];  // [512][256][256]
  const float* bias       = (const float*)d_in[3];  // [512]
  const float* gamma      = (const float*)d_in[4];  // [512]
  const float* beta       = (const float*)d_in[5];  // [512]
  const int*   sequence   = (const int*)d_in[6];    // [16][512]
  const int*   king_id    = (const int*)d_in[7];    // [16][1]

  float* out = (float*)d_out;                       // [16][512][512]
  float* T   = (float*)d_ws;                        // [16][256][512] = 8 MB

  // Stage 1: 8192 wave-tiles / 8 waves per block.
  king_w_gemm<<<1024, 256, 0, stream>>>(Wm, king_table, king_id, T);
  // Stage 2: one block per (batch, s-tile).
  seq_t_gemm<<<512, 256, 0, stream>>>(tables, sequence, king_id, T, bias, out);
  // Stage 3: 8192 rows / 8 waves per block.
  layernorm_inplace<<<1024, 256, 0, stream>>>(out, gamma, beta);
}